// MultiQueryAttention_56169582297273
// MI455X (gfx1250) — compile-verified
//
#include <hip/hip_runtime.h>
#include <hip/hip_bf16.h>

// MQA block for MI455X (gfx1250, wave32), fp32 end-to-end using the native
// fp32 WMMA path V_WMMA_F32_16X16X4_F32.
//
// Pipeline:
//   1) qkv = x @ w_attn            (WMMA GEMM, 8192x2304x2048)
//   2) RoPE on q (16 heads) and k  (elementwise)
//   3) flash attention per (b,h)   (WMMA for QK^T and PV, online softmax)
//   4) out = y @ w_proj            (WMMA GEMM, 8192x2048x2048)
//
// All LDS operand tiles are laid out so that every WMMA fragment load is a
// contiguous 8-byte (v2f) LDS access -> single ds_load_b64 per fragment.
// GEMM uses a 64x64 wave tile (4x4 accumulators) so each k-step needs only
// 8 fragment loads for 16 WMMAs.

typedef float v2f __attribute__((ext_vector_type(2)));
typedef float v8f __attribute__((ext_vector_type(8)));

__device__ __forceinline__ v8f wmma_f32(v2f a, v2f b, v8f c) {
  // 8 args: (neg_a, A, neg_b, B, c_mod, C, reuse_a, reuse_b)
  return __builtin_amdgcn_wmma_f32_16x16x4_f32(false, a, false, b, (short)0, c,
                                               false, false);
}

__device__ __forceinline__ v8f v8f_zero() {
  v8f v;
#pragma unroll
  for (int i = 0; i < 8; ++i) v[i] = 0.0f;
  return v;
}

// ---------------------------------------------------------------------------
// Generic fp32 WMMA GEMM: C[M,N] = A[M,K] * B[K,N], all row-major.
// Block tile 256x128, K-tile 32. 8 waves (wave32) in 4(M) x 2(N) grid,
// each wave owns a 64x64 tile = 4x4 accumulators of 16x16.
// B is staged TRANSPOSED in LDS (Bs[col][k]) so B fragments are contiguous.
// Requires M%256==0, N%128==0, K%32==0 (true for all shapes here).
// ---------------------------------------------------------------------------
#define GBM 256
#define GBN 128
#define GBK 32

__global__ __launch_bounds__(256) void gemm_f32_wmma(
    const float* __restrict__ A, const float* __restrict__ B,
    float* __restrict__ C, int M, int N, int K) {
  __shared__ __align__(16) float As[GBM][GBK + 4];  // 256 x 36 (36.0 KB)
  __shared__ __align__(16) float Bs[GBN][GBK + 4];  // 128 x 36 (18.0 KB), B^T

  const int tid = threadIdx.x;
  const int lane = tid & 31;
  const int wave = tid >> 5;  // 0..7
  const int wm = wave >> 1;   // 0..3  : 64 rows each
  const int wn = wave & 1;    // 0..1  : 64 cols each
  const int lhalf = lane >> 4;
  const int l16 = lane & 15;

  const int block_m = blockIdx.y * GBM;
  const int block_n = blockIdx.x * GBN;

  v8f acc[4][4];
#pragma unroll
  for (int mi = 0; mi < 4; ++mi)
#pragma unroll
    for (int ni = 0; ni < 4; ++ni) acc[mi][ni] = v8f_zero();

  for (int k0 = 0; k0 < K; k0 += GBK) {
    // Stage A tile: 256 rows x 32 cols = 2048 float4, 8 per thread.
#pragma unroll
    for (int i = 0; i < 8; ++i) {
      int idx = tid + i * 256;
      int row = idx >> 3;  // 8 float4 per row
      int c4 = idx & 7;
      float4 v = *(const float4*)&A[(size_t)(block_m + row) * K + k0 + c4 * 4];
      *(float4*)&As[row][c4 * 4] = v;
    }
    // Stage B tile transposed: read [k][n] coalesced, write Bs[n][k].
#pragma unroll
    for (int i = 0; i < 4; ++i) {
      int idx = tid + i * 256;
      int krow = idx >> 5;  // 32 float4 per k-row
      int c4 = idx & 31;
      float4 v = *(const float4*)&B[(size_t)(k0 + krow) * N + block_n + c4 * 4];
      Bs[c4 * 4 + 0][krow] = v.x;
      Bs[c4 * 4 + 1][krow] = v.y;
      Bs[c4 * 4 + 2][krow] = v.z;
      Bs[c4 * 4 + 3][krow] = v.w;
    }
    // Prefetch next K-tile, one 128B cacheline per thread (global_prefetch_b8).
    if (k0 + GBK < K) {
      // A: 256 rows x 128B (one k-tile row == one cacheline)
      __builtin_prefetch(&A[(size_t)(block_m + tid) * K + k0 + GBK], 0, 0);
      // B: 32 rows x 512B -> 128 lines, threads 0..127
      if (tid < 128) {
        __builtin_prefetch(
            &B[(size_t)(k0 + GBK + (tid >> 2)) * N + block_n + (tid & 3) * 32],
            0, 0);
      }
    }
    __syncthreads();

#pragma unroll
    for (int kk = 0; kk < GBK; kk += 4) {
      const int ka = kk + lhalf * 2;  // this lane-half's K pair
      v2f afrag[4], bfrag[4];
#pragma unroll
      for (int mi = 0; mi < 4; ++mi) {
        int row = wm * 64 + mi * 16 + l16;
        afrag[mi] = *(const v2f*)&As[row][ka];
      }
#pragma unroll
      for (int ni = 0; ni < 4; ++ni) {
        int col = wn * 64 + ni * 16 + l16;
        bfrag[ni] = *(const v2f*)&Bs[col][ka];
      }
#pragma unroll
      for (int mi = 0; mi < 4; ++mi)
#pragma unroll
        for (int ni = 0; ni < 4; ++ni)
          acc[mi][ni] = wmma_f32(afrag[mi], bfrag[ni], acc[mi][ni]);
    }
    __syncthreads();
  }

  // Store: C/D layout — VGPR r holds row (r, r+8) halves, lanes are columns.
#pragma unroll
  for (int mi = 0; mi < 4; ++mi)
#pragma unroll
    for (int ni = 0; ni < 4; ++ni)
#pragma unroll
      for (int r = 0; r < 8; ++r) {
        int row = block_m + wm * 64 + mi * 16 + r + lhalf * 8;
        int col = block_n + wn * 64 + ni * 16 + l16;
        C[(size_t)row * N + col] = acc[mi][ni][r];
      }
}

// ---------------------------------------------------------------------------
// RoPE applied in-place to qkv[B*T][2304]: 16 q heads + 1 shared k head.
// One thread per (bt, head 0..16, d 0..63); head 16 maps to the k slice
// (offset 16*128 == 2048).
// ---------------------------------------------------------------------------
__global__ void rope_kernel(float* __restrict__ qkv, int total, int T) {
  int idx = blockIdx.x * blockDim.x + threadIdx.x;
  if (idx >= total) return;
  const int d = idx & 63;
  const int h = (idx >> 6) % 17;
  const int bt = idx / (64 * 17);
  const int t = bt % T;

  const float log_base_over = 9.210340371976184f / 64.0f;  // ln(10000)/64
  const float inv_freq = __expf(-(float)d * log_base_over);
  const float f = (float)t * inv_freq;
  float s, c;
  __sincosf(f, &s, &c);

  float* base = qkv + (size_t)bt * 2304 + h * 128;
  const float x1 = base[d];
  const float x2 = base[d + 64];
  base[d] = x1 * c - x2 * s;
  base[d + 64] = x2 * c + x1 * s;
}

// ---------------------------------------------------------------------------
// Flash attention (non-causal, multi-query). One block per (b, h, 64-row Q
// tile). Streams 32-key K/V blocks through a single shared buffer:
//   - K staged as Ks[key][d]   (QK^T contraction over d is contiguous)
//   - V staged as Vt[d][key]   (PV contraction over key is contiguous)
// Online softmax state (m, l) per q-row in LDS. All matmuls via fp32 WMMA.
// Writes y in (B, T, H, D) layout -> row-major [B*T, 2048] for the proj GEMM.
// ---------------------------------------------------------------------------
#define FQ 64
#define FKV 32
#define HD 128

#define KS(key, d) KVraw[(key) * (HD + 4) + (d)]
#define VT(d, key) KVraw[(d) * (FKV + 4) + (key)]

__global__ __launch_bounds__(256) void mqa_flash(const float* __restrict__ qkv,
                                                 float* __restrict__ y, int T) {
  __shared__ __align__(16) float Qs[FQ][HD + 4];        // 33.0 KB
  __shared__ __align__(16) float KVraw[HD * (FKV + 4)]; // 18.0 KB (K or V^T)
  __shared__ __align__(16) float Ss[FQ][FKV + 4];       //  9.0 KB
  __shared__ float m_s[FQ], l_s[FQ], fac_s[FQ];

  const int tid = threadIdx.x;
  const int lane = tid & 31;
  const int wave = tid >> 5;
  const int lhalf = lane >> 4;
  const int l16 = lane & 15;

  const int q0 = blockIdx.x * FQ;
  const int bh = blockIdx.y;
  const int b = bh >> 4;
  const int h = bh & 15;

  const float scale = 0.08838834764831845f;  // 1/sqrt(128)
  const size_t rs = 2304;                    // qkv row stride
  const float* qbase = qkv + (size_t)b * T * rs + (size_t)h * 128;
  const float* kbase = qkv + (size_t)b * T * rs + 2048;
  const float* vbase = kbase + 128;

  // Load Q tile: 64 x 128 = 2048 float4, 8 per thread.
#pragma unroll
  for (int i = 0; i < 8; ++i) {
    int idx = tid + i * 256;
    int row = idx >> 5;
    int c4 = idx & 31;
    *(float4*)&Qs[row][c4 * 4] =
        *(const float4*)&qbase[(size_t)(q0 + row) * rs + c4 * 4];
  }
  if (tid < FQ) {
    m_s[tid] = -1e30f;
    l_s[tid] = 0.0f;
  }

  // S wave grid: 4(M rows of 16) x 2(N keys of 16) covers 64x32.
  const int wmS = wave >> 1;
  const int wnS = wave & 1;
  // O wave grid: 2(M rows of 32) x 4(N dims of 32) covers 64x128.
  const int wmO = wave >> 2;
  const int wnO = wave & 3;

  v8f o[2][2];
#pragma unroll
  for (int mi = 0; mi < 2; ++mi)
#pragma unroll
    for (int ni = 0; ni < 2; ++ni) o[mi][ni] = v8f_zero();

  __syncthreads();

  for (int k0 = 0; k0 < T; k0 += FKV) {
    // ---- Stage K block as Ks[key][d] (32 x 128) ----
#pragma unroll
    for (int i = 0; i < 4; ++i) {
      int idx = tid + i * 256;
      int row = idx >> 5;
      int c4 = idx & 31;
      *(float4*)&KS(row, c4 * 4) =
          *(const float4*)&kbase[(size_t)(k0 + row) * rs + c4 * 4];
    }
    // Prefetch the V block we'll need right after (global_prefetch_b8).
    __builtin_prefetch(
        &vbase[(size_t)(k0 + (tid >> 3)) * rs + (tid & 7) * 16], 0, 0);
    __syncthreads();

    // ---- S = Q K^T over D=128 (32 WMMA k-steps), one 16x16 per wave ----
    v8f sacc = v8f_zero();
#pragma unroll
    for (int kk = 0; kk < HD; kk += 4) {
      const int ka = kk + lhalf * 2;
      const int qrow = wmS * 16 + l16;
      const int key = wnS * 16 + l16;
      v2f a = *(const v2f*)&Qs[qrow][ka];
      v2f bfr = *(const v2f*)&KS(key, ka);  // B[k_dim][key] = K[key][k_dim]
      sacc = wmma_f32(a, bfr, sacc);
    }
#pragma unroll
    for (int r = 0; r < 8; ++r) {
      int row = wmS * 16 + r + lhalf * 8;
      int col = wnS * 16 + l16;
      Ss[row][col] = sacc[r] * scale;
    }
    __syncthreads();

    // ---- Stage V transposed as Vt[d][key] (K reads done) + softmax ----
#pragma unroll
    for (int i = 0; i < 4; ++i) {
      int idx = tid + i * 256;
      int row = idx >> 5;  // key
      int c4 = idx & 31;   // d group
      float4 v = *(const float4*)&vbase[(size_t)(k0 + row) * rs + c4 * 4];
      VT(c4 * 4 + 0, row) = v.x;
      VT(c4 * 4 + 1, row) = v.y;
      VT(c4 * 4 + 2, row) = v.z;
      VT(c4 * 4 + 3, row) = v.w;
    }
    if (tid < FQ) {
      float mold = m_s[tid];
      float bm = -1e30f;
      float srow[FKV];
#pragma unroll
      for (int j = 0; j < FKV; ++j) {
        srow[j] = Ss[tid][j];
        bm = fmaxf(bm, srow[j]);
      }
      const float mnew = fmaxf(mold, bm);
      const float fac = __expf(mold - mnew);
      float lsum = 0.0f;
#pragma unroll
      for (int j = 0; j < FKV; ++j) {
        float p = __expf(srow[j] - mnew);
        Ss[tid][j] = p;
        lsum += p;
      }
      m_s[tid] = mnew;
      l_s[tid] = l_s[tid] * fac + lsum;
      fac_s[tid] = fac;
    }
    __syncthreads();

    // ---- Rescale running O, then O += P @ V (32-key contraction) ----
#pragma unroll
    for (int mi = 0; mi < 2; ++mi)
#pragma unroll
      for (int r = 0; r < 8; ++r) {
        int row = wmO * 32 + mi * 16 + r + lhalf * 8;
        float f = fac_s[row];
        o[mi][0][r] *= f;
        o[mi][1][r] *= f;
      }
#pragma unroll
    for (int kk = 0; kk < FKV; kk += 4) {
      const int ka = kk + lhalf * 2;
      v2f a[2], bb[2];
#pragma unroll
      for (int mi = 0; mi < 2; ++mi) {
        int row = wmO * 32 + mi * 16 + l16;
        a[mi] = *(const v2f*)&Ss[row][ka];
      }
#pragma unroll
      for (int ni = 0; ni < 2; ++ni) {
        int col = wnO * 32 + ni * 16 + l16;
        bb[ni] = *(const v2f*)&VT(col, ka);  // V[key][d] -> Vt[d][key]
      }
#pragma unroll
      for (int mi = 0; mi < 2; ++mi)
#pragma unroll
        for (int ni = 0; ni < 2; ++ni)
          o[mi][ni] = wmma_f32(a[mi], bb[ni], o[mi][ni]);
    }
    __syncthreads();  // protect KVraw/Ss before next iteration's stores
  }

  // ---- Epilogue: normalize by l and store y in (B,T,H,D) ----
#pragma unroll
  for (int mi = 0; mi < 2; ++mi)
#pragma unroll
    for (int ni = 0; ni < 2; ++ni)
#pragma unroll
      for (int r = 0; r < 8; ++r) {
        int row = wmO * 32 + mi * 16 + r + lhalf * 8;
        int d = wnO * 32 + ni * 16 + l16;
        float val = o[mi][ni][r] / l_s[row];
        y[((size_t)b * T + q0 + row) * 2048 + (size_t)h * 128 + d] = val;
      }
}

// ---------------------------------------------------------------------------
extern "C" void kernel_launch(void* const* d_in, const int* in_sizes, int n_in,
                              void* d_out, int out_size, void* d_ws,
                              size_t ws_size, hipStream_t stream) {
  (void)in_sizes;
  (void)n_in;
  (void)out_size;
  (void)ws_size;

  const float* x = (const float*)d_in[0];       // [4,2048,2048]
  const float* w_attn = (const float*)d_in[1];  // [2048,2304]
  const float* w_proj = (const float*)d_in[2];  // [2048,2048]
  float* out = (float*)d_out;                   // [4,2048,2048]

  const int B = 4, T = 2048, C = 2048;
  const int M = B * T;    // 8192
  const int Nqkv = 2304;  // C + 2*HEAD_DIM

  float* qkv = (float*)d_ws;          // 8192*2304 f32 (75.5 MB)
  float* y = qkv + (size_t)M * Nqkv;  // 8192*2048 f32 (64 MB)

  // 1) qkv = x @ w_attn
  gemm_f32_wmma<<<dim3(Nqkv / GBN, M / GBM), dim3(256), 0, stream>>>(
      x, w_attn, qkv, M, Nqkv, C);

  // 2) RoPE on q heads and shared k
  {
    int total = B * T * 17 * 64;
    rope_kernel<<<(total + 255) / 256, 256, 0, stream>>>(qkv, total, T);
  }

  // 3) flash MQA -> y in [B*T, 2048]
  mqa_flash<<<dim3(T / FQ, B * 16), dim3(256), 0, stream>>>(qkv, y, T);

  // 4) out = y @ w_proj
  gemm_f32_wmma<<<dim3(C / GBN, M / GBM), dim3(256), 0, stream>>>(
      y, w_proj, out, M, C, C);
}